// BasicRNN_42271068127787
// MI455X (gfx1250) — compile-verified
//
#include <hip/hip_runtime.h>
#include <math.h>

#define HIDDEN  6
#define INPUT   20
#define SEQ     34
#define CLASSES 20
#define FEAT    (SEQ * HIDDEN)   // 204
#define GPAIRS  (2 * HIDDEN)     // 12 gate pairs (24 gates)

typedef float v2f __attribute__((ext_vector_type(2)));
typedef float v8f __attribute__((ext_vector_type(8)));

// ---- packed-weight layout inside d_ws (offsets in floats, after feat buffer) ----
#define PK_W0 0
#define PK_H0 (PK_W0 + INPUT * 24)    // 480
#define PK_B0 (PK_H0 + HIDDEN * 24)   // 624
#define PK_W1 (PK_B0 + 24)            // 648
#define PK_H1 (PK_W1 + HIDDEN * 24)   // 792
#define PK_B1 (PK_H1 + HIDDEN * 24)   // 936
#define PK_W2 (PK_B1 + 24)            // 960
#define PK_H2 (PK_W2 + HIDDEN * 24)   // 1104
#define PK_B2 (PK_H2 + HIDDEN * 24)   // 1248
#define PK_TOTAL (PK_B2 + 24)         // 1272 floats

__device__ __forceinline__ float fsig(float x) {
    return 1.0f / (1.0f + __expf(-x));
}
// tanh(x) = 1 - 2/(exp(2x)+1); correct saturation at +/-inf
__device__ __forceinline__ float ftanh_fast(float x) {
    return 1.0f - 2.0f / (__expf(2.0f * x) + 1.0f);
}

// Packed dual-FMA on float2 -> v_pk_fma_f32 (compiler-selected VOP3P;
// smaller than the scalar form: 7628 vs 9916 static VALU).
__device__ __forceinline__ void pk_fma(v2f& acc, v2f w, v2f x) {
    acc = __builtin_elementwise_fma(w, x, acc);
}

// ---------------------------------------------------------------------------
// One-shot weight repack: gate-pair-major so the recurrence runs on
// v_pk_fma_f32. pk_w[i*12+p] = { w[2p][i], w[2p+1][i] }, bias pre-summed.
// ---------------------------------------------------------------------------
__device__ __forceinline__ void pack_layer(const float* __restrict__ wih,
                                           const float* __restrict__ whh,
                                           const float* __restrict__ bih,
                                           const float* __restrict__ bhh,
                                           float* wP, float* hP, float* bP,
                                           int IN, int p) {
    for (int i = 0; i < IN; ++i) {
        wP[(i * GPAIRS + p) * 2 + 0] = wih[(2 * p + 0) * IN + i];
        wP[(i * GPAIRS + p) * 2 + 1] = wih[(2 * p + 1) * IN + i];
    }
    for (int i = 0; i < HIDDEN; ++i) {
        hP[(i * GPAIRS + p) * 2 + 0] = whh[(2 * p + 0) * HIDDEN + i];
        hP[(i * GPAIRS + p) * 2 + 1] = whh[(2 * p + 1) * HIDDEN + i];
    }
    bP[2 * p + 0] = bih[2 * p + 0] + bhh[2 * p + 0];
    bP[2 * p + 1] = bih[2 * p + 1] + bhh[2 * p + 1];
}

__global__ void pack_weights_kernel(
    const float* __restrict__ wih0, const float* __restrict__ whh0,
    const float* __restrict__ bih0, const float* __restrict__ bhh0,
    const float* __restrict__ wih1, const float* __restrict__ whh1,
    const float* __restrict__ bih1, const float* __restrict__ bhh1,
    const float* __restrict__ wih2, const float* __restrict__ whh2,
    const float* __restrict__ bih2, const float* __restrict__ bhh2,
    float* __restrict__ pk) {
    const int p = threadIdx.x;
    if (p >= GPAIRS) return;
    pack_layer(wih0, whh0, bih0, bhh0, pk + PK_W0, pk + PK_H0, pk + PK_B0, INPUT, p);
    pack_layer(wih1, whh1, bih1, bhh1, pk + PK_W1, pk + PK_H1, pk + PK_B1, HIDDEN, p);
    pack_layer(wih2, whh2, bih2, bhh2, pk + PK_W2, pk + PK_H2, pk + PK_B2, HIDDEN, p);
}

// ---------------------------------------------------------------------------
// LSTM cell step, one batch element per lane, gate pairs on packed f32 FMA.
// Weights are wave-uniform (s_load); x/h broadcast per lane.
// ---------------------------------------------------------------------------
template <int IN>
__device__ __forceinline__ void lstm_step_pk(const v2f* __restrict__ wP,  // [IN][12]
                                             const v2f* __restrict__ hP,  // [6][12]
                                             const v2f* __restrict__ bP,  // [12]
                                             const float* xin, float* h, float* c) {
    v2f g2[GPAIRS];
#pragma unroll
    for (int p = 0; p < GPAIRS; ++p) g2[p] = bP[p];
#pragma unroll
    for (int i = 0; i < IN; ++i) {
        const v2f xi = {xin[i], xin[i]};
#pragma unroll
        for (int p = 0; p < GPAIRS; ++p) pk_fma(g2[p], wP[i * GPAIRS + p], xi);
    }
#pragma unroll
    for (int i = 0; i < HIDDEN; ++i) {
        const v2f hi = {h[i], h[i]};
#pragma unroll
        for (int p = 0; p < GPAIRS; ++p) pk_fma(g2[p], hP[i * GPAIRS + p], hi);
    }
    float g[4 * HIDDEN];
#pragma unroll
    for (int p = 0; p < GPAIRS; ++p) {
        g[2 * p + 0] = g2[p].x;
        g[2 * p + 1] = g2[p].y;
    }
    // torch gate order: i, f, g, o
#pragma unroll
    for (int j = 0; j < HIDDEN; ++j) {
        float ig = fsig(g[j]);
        float fg = fsig(g[HIDDEN + j]);
        float gg = ftanh_fast(g[2 * HIDDEN + j]);
        float og = fsig(g[3 * HIDDEN + j]);
        float cn = fg * c[j] + ig * gg;
        c[j] = cn;
        h[j] = og * ftanh_fast(cn);
    }
}

__global__ __launch_bounds__(128) void lstm_kernel(
    const float* __restrict__ x, const float* __restrict__ pk,
    float* __restrict__ feat, int B) {
    const int b = blockIdx.x * blockDim.x + threadIdx.x;
    if (b >= B) return;
    const float* xb = x + (size_t)b * (INPUT * SEQ);  // x[b][i][t] = xb[i*SEQ + t]
    float* fb = feat + (size_t)b * FEAT;

    const v2f* w0 = (const v2f*)(pk + PK_W0);
    const v2f* hh0 = (const v2f*)(pk + PK_H0);
    const v2f* bb0 = (const v2f*)(pk + PK_B0);
    const v2f* w1 = (const v2f*)(pk + PK_W1);
    const v2f* hh1 = (const v2f*)(pk + PK_H1);
    const v2f* bb1 = (const v2f*)(pk + PK_B1);
    const v2f* w2 = (const v2f*)(pk + PK_W2);
    const v2f* hh2 = (const v2f*)(pk + PK_H2);
    const v2f* bb2 = (const v2f*)(pk + PK_B2);

    float h0[HIDDEN], c0[HIDDEN], h1[HIDDEN], c1[HIDDEN], h2[HIDDEN], c2[HIDDEN];
#pragma unroll
    for (int j = 0; j < HIDDEN; ++j) {
        h0[j] = c0[j] = h1[j] = c1[j] = h2[j] = c2[j] = 0.0f;
    }

#pragma unroll 1  // keep I$ footprint small; body already has 2-step ILP
    for (int t = 0; t < SEQ; t += 2) {  // t even -> 8B-aligned pair loads
        float xa[INPUT], xc[INPUT];
#pragma unroll
        for (int i = 0; i < INPUT; ++i) {
            const v2f p2 = *(const v2f*)(xb + i * SEQ + t);  // x[.,i,t], x[.,i,t+1]
            xa[i] = p2.x;
            xc[i] = p2.y;
        }
#pragma unroll
        for (int u = 0; u < 2; ++u) {
            const float* xt = (u == 0) ? xa : xc;
            lstm_step_pk<INPUT>(w0, hh0, bb0, xt, h0, c0);
            lstm_step_pk<HIDDEN>(w1, hh1, bb1, h0, h1, c1);
            lstm_step_pk<HIDDEN>(w2, hh2, bb2, h1, h2, c2);
#pragma unroll
            for (int p = 0; p < HIDDEN / 2; ++p) {  // 3 aligned b64 stores
                v2f s2 = {h2[2 * p], h2[2 * p + 1]};
                *(v2f*)(fb + (t + u) * HIDDEN + 2 * p) = s2;
            }
        }
    }
}

// ---------------------------------------------------------------------------
// FC (B x 204) @ (204 x 20) via V_WMMA_F32_16X16X4_F32 + fused log_softmax.
// One wave handles 32 batch rows (two M-tiles sharing the B-matrix loads).
// ---------------------------------------------------------------------------
__global__ __launch_bounds__(32) void fc_softmax_kernel(
    const float* __restrict__ feat,
    const float* __restrict__ fcw,  // (20, 204)
    const float* __restrict__ fcb,  // (20,)
    float* __restrict__ out,        // (B, 20) log-probs
    int B) {
    const int lane = threadIdx.x;   // 0..31
    const int b0 = blockIdx.x * 32;
    const int m  = lane & 15;       // A row-in-tile / B column
    const int kh = lane >> 4;       // which K pair this half-wave supplies

    const float* arow0 = feat + (size_t)(b0 + m) * FEAT;        // rows 0..15
    const float* arow1 = arow0 + (size_t)16 * FEAT;             // rows 16..31
    const float* brow0 = fcw + m * FEAT;                        // classes 0..15
    const int   cls1   = (m < 4) ? (16 + m) : 0;
    const float* brow1 = fcw + cls1 * FEAT;
    const v2f   maskv  = {(m < 4) ? 1.0f : 0.0f, (m < 4) ? 1.0f : 0.0f};

    v8f a00 = {}, a01 = {}, a10 = {}, a11 = {};  // [Mtile][Ntile]

    for (int k0 = 0; k0 < FEAT; k0 += 4) {
        const int ka = k0 + 2 * kh;                  // even -> aligned b64 loads
        const v2f av0 = *(const v2f*)(arow0 + ka);
        const v2f av1 = *(const v2f*)(arow1 + ka);
        const v2f bv0 = *(const v2f*)(brow0 + ka);
        const v2f bv1 = *(const v2f*)(brow1 + ka) * maskv;  // zero pad cls>=20
        a00 = __builtin_amdgcn_wmma_f32_16x16x4_f32(false, av0, false, bv0,
                                                    (short)0, a00, false, false);
        a01 = __builtin_amdgcn_wmma_f32_16x16x4_f32(false, av0, false, bv1,
                                                    (short)0, a01, false, false);
        a10 = __builtin_amdgcn_wmma_f32_16x16x4_f32(false, av1, false, bv0,
                                                    (short)0, a10, false, false);
        a11 = __builtin_amdgcn_wmma_f32_16x16x4_f32(false, av1, false, bv1,
                                                    (short)0, a11, false, false);
    }

    // C/D layout: lane l, VGPR r -> (M = r + 8*(l>>4), N = l&15)
    __shared__ float sm[32][24];
#pragma unroll
    for (int r = 0; r < 8; ++r) {
        sm[r + 8 * kh][m]      = a00[r];
        sm[16 + r + 8 * kh][m] = a10[r];
    }
    if (m < 4) {
#pragma unroll
        for (int r = 0; r < 8; ++r) {
            sm[r + 8 * kh][16 + m]      = a01[r];
            sm[16 + r + 8 * kh][16 + m] = a11[r];
        }
    }
    __syncthreads();

    // log_softmax: each lane owns one of the 32 rows
    const int row = b0 + lane;
    float v[CLASSES];
    float mx = -INFINITY;
#pragma unroll
    for (int c = 0; c < CLASSES; ++c) {
        v[c] = sm[lane][c] + fcb[c];
        mx = fmaxf(mx, v[c]);
    }
    float s = 0.0f;
#pragma unroll
    for (int c = 0; c < CLASSES; ++c) s += __expf(v[c] - mx);
    const float lse = __logf(s);
    float* o = out + (size_t)row * CLASSES;
#pragma unroll
    for (int c = 0; c < CLASSES; ++c) o[c] = v[c] - mx - lse;
}

extern "C" void kernel_launch(void* const* d_in, const int* in_sizes, int n_in,
                              void* d_out, int out_size, void* d_ws, size_t ws_size,
                              hipStream_t stream) {
    const float* x    = (const float*)d_in[0];
    const float* wih0 = (const float*)d_in[1];
    const float* whh0 = (const float*)d_in[2];
    const float* bih0 = (const float*)d_in[3];
    const float* bhh0 = (const float*)d_in[4];
    const float* wih1 = (const float*)d_in[5];
    const float* whh1 = (const float*)d_in[6];
    const float* bih1 = (const float*)d_in[7];
    const float* bhh1 = (const float*)d_in[8];
    const float* wih2 = (const float*)d_in[9];
    const float* whh2 = (const float*)d_in[10];
    const float* bih2 = (const float*)d_in[11];
    const float* bhh2 = (const float*)d_in[12];
    const float* fcw  = (const float*)d_in[13];
    const float* fcb  = (const float*)d_in[14];

    const int B = in_sizes[0] / (INPUT * SEQ);  // 32768
    float* feat = (float*)d_ws;                 // B * 204 floats (~26.7 MB)
    float* pk   = feat + (size_t)B * FEAT;      // + 1272 floats packed weights
    float* out  = (float*)d_out;

    pack_weights_kernel<<<1, 32, 0, stream>>>(
        wih0, whh0, bih0, bhh0, wih1, whh1, bih1, bhh1,
        wih2, whh2, bih2, bhh2, pk);

    lstm_kernel<<<(B + 127) / 128, 128, 0, stream>>>(x, pk, feat, B);

    fc_softmax_kernel<<<B / 32, 32, 0, stream>>>(feat, fcw, fcb, out, B);
}